// GraphAttentionLayer_25348896981141
// MI455X (gfx1250) — compile-verified
//
#include <hip/hip_runtime.h>
#include <math.h>

// GAT layer for MI455X (gfx1250, wave32).
//  1) Wh = h @ W                (WMMA f32 16x16x4 GEMM, software-pipelined)
//  2) e1 = Wh@a1, e2 = Wh@a2    (wave-per-row reduction)
//  3) attention = softmax(mask(leakyrelu(e1_i + e2_j)))  -> d_out[att]
//  4) h' = elu(attention @ Wh)  (same WMMA GEMM)         -> d_out[h']

typedef __attribute__((ext_vector_type(2))) float v2f;
typedef __attribute__((ext_vector_type(8))) float v8f;

#define LRELU_ALPHA 0.2f
#define GAT_NEG_INF -9.0e15f
#define NT 16      // 16 col-tiles of 16 => 256 columns per wave strip
#define WAVES 8    // waves per 256-thread block

// Load one k-step's B fragments (16 tiles): b[t] = { B[k+2h][16t+n], B[k+2h+1][16t+n] }
__device__ __forceinline__ void load_bfrag(const float* __restrict__ bp, int ldb,
                                           v2f* __restrict__ bf)
{
#pragma unroll
    for (int t = 0; t < NT; ++t) {
        bf[t].x = bp[t * 16];
        bf[t].y = bp[t * 16 + ldb];
    }
}

__device__ __forceinline__ void mma_chain(v2f a, const v2f* __restrict__ bf,
                                          v8f* __restrict__ acc)
{
#pragma unroll
    for (int t = 0; t < NT; ++t) {
        acc[t] = __builtin_amdgcn_wmma_f32_16x16x4_f32(
            false, a, false, bf[t], (short)0, acc[t], false, false);
    }
}

// ---------------------------------------------------------------------------
// WMMA fp32 GEMM: C[M,256] = A[M,K] * B[K,256] (+ optional ELU epilogue).
// One wave owns a 16-row x 256-col strip of C (16 v8f accumulators).
// A (large, streaming) is read exactly once from HBM; B is L2-resident.
// Register double-buffered over k-steps of 4 so loads for step k+4 overlap
// the 16-WMMA chain of step k.
//
// Fragment layouts (CDNA5 ISA 7.12.2, 32-bit matrices):
//   A 16x4:  (m,k) -> lane = m + 16*(k>>1), vgpr = k&1
//   B 4x16:  (k,n) -> lane = n + 16*(k>>1), vgpr = k&1
//   C 16x16: (m,n) -> lane = n + 16*(m>>3), vgpr = m&7
// ---------------------------------------------------------------------------
__global__ __launch_bounds__(256)
void gat_gemm_wmma(const float* __restrict__ A, const float* __restrict__ B,
                   float* __restrict__ C, int M, int K,
                   int lda, int ldb, int ldc,
                   long bsA, long bsB, long bsC, int doElu)
{
    const int lane   = threadIdx.x & 31;
    const int wave   = threadIdx.x >> 5;
    const int laneLo = lane & 15;
    const int laneHi = lane >> 4;          // 0: K pair {0,1}, 1: K pair {2,3}

    const int strip = blockIdx.x * WAVES + wave;
    const int row0  = strip * 16;
    if (row0 >= M) return;                 // wave-uniform

    const float* Ab = A + (long)blockIdx.y * bsA;
    const float* Bb = B + (long)blockIdx.y * bsB;
    float*       Cb = C + (long)blockIdx.y * bsC;

    const float* aptr  = Ab + (long)(row0 + laneLo) * lda + 2 * laneHi;
    const float* bbase = Bb + (long)(2 * laneHi) * ldb + laneLo;

    v8f zero = {};
    v8f acc[NT];
#pragma unroll
    for (int t = 0; t < NT; ++t) acc[t] = zero;

    // Software pipeline: (a0,b0) holds fragments for step k at loop top.
    v2f a0, a1;
    v2f b0[NT], b1[NT];
    a0 = *(const v2f*)(aptr);
    load_bfrag(bbase, ldb, b0);

    int k = 0;
    for (; k < K - 8; k += 8) {
        if ((k & 63) == 0) __builtin_prefetch(aptr + k + 128, 0, 3);
        a1 = *(const v2f*)(aptr + k + 4);
        load_bfrag(bbase + (long)(k + 4) * ldb, ldb, b1);
        mma_chain(a0, b0, acc);
        a0 = *(const v2f*)(aptr + k + 8);
        load_bfrag(bbase + (long)(k + 8) * ldb, ldb, b0);
        mma_chain(a1, b1, acc);
    }
    // k == K-8: two steps remain (K-8 in a0/b0, then K-4)
    a1 = *(const v2f*)(aptr + k + 4);
    load_bfrag(bbase + (long)(k + 4) * ldb, ldb, b1);
    mma_chain(a0, b0, acc);
    mma_chain(a1, b1, acc);

    // Store: acc[t][r] -> C[row0 + r + 8*laneHi][16*t + laneLo]
#pragma unroll
    for (int t = 0; t < NT; ++t) {
        float* cp = Cb + (long)(row0 + 8 * laneHi) * ldc + t * 16 + laneLo;
#pragma unroll
        for (int r = 0; r < 8; ++r) {
            float v = acc[t][r];
            if (doElu) v = (v > 0.f) ? v : (__expf(v) - 1.f);
            cp[(long)r * ldc] = v;
        }
    }
}

// ---------------------------------------------------------------------------
// e1[row] = Wh[row,:] . a[0:F], e2[row] = Wh[row,:] . a[F:2F]; one wave per row
// ---------------------------------------------------------------------------
__global__ __launch_bounds__(256)
void gat_edge(const float* __restrict__ Wh, const float* __restrict__ a,
              float* __restrict__ e1, float* __restrict__ e2, int F)
{
    const int lane = threadIdx.x & 31;
    const int row  = blockIdx.x * WAVES + (threadIdx.x >> 5);
    const float* wh = Wh + (long)row * F;
    float s1 = 0.f, s2 = 0.f;
    for (int f = lane; f < F; f += 32) {
        float v = wh[f];
        s1 += v * a[f];
        s2 += v * a[F + f];
    }
#pragma unroll
    for (int off = 16; off > 0; off >>= 1) {
        s1 += __shfl_xor(s1, off, 32);
        s2 += __shfl_xor(s2, off, 32);
    }
    if (lane == 0) { e1[row] = s1; e2[row] = s2; }
}

// ---------------------------------------------------------------------------
// attention[b,i,:] = softmax_j( adj>0 ? leakyrelu(e1[b,i]+e2[b,j]) : NEG_INF )
// One 256-thread block per row; 8 row elements per thread in registers.
// ---------------------------------------------------------------------------
__global__ __launch_bounds__(256)
void gat_softmax(const int* __restrict__ adj, const float* __restrict__ e1,
                 const float* __restrict__ e2, float* __restrict__ att, int N)
{
    const int b    = blockIdx.y;
    const int i    = blockIdx.x;
    const int tid  = threadIdx.x;
    const int lane = tid & 31;
    const int wave = tid >> 5;
    __shared__ float sred[WAVES];

    const long rowBase = ((long)b * N + i) * (long)N;
    const int*   adjrow = adj + rowBase;
    const float* e2b    = e2 + (long)b * N;
    const float  e1v    = e1[(long)b * N + i];

    float vals[8];
    float mx = -INFINITY;
#pragma unroll
    for (int u = 0; u < 8; ++u) {
        int j = tid + u * 256;
        float ev = e1v + e2b[j];
        ev = (ev > 0.f) ? ev : (LRELU_ALPHA * ev);
        ev = (adjrow[j] > 0) ? ev : GAT_NEG_INF;
        vals[u] = ev;
        mx = fmaxf(mx, ev);
    }
#pragma unroll
    for (int off = 16; off > 0; off >>= 1) mx = fmaxf(mx, __shfl_xor(mx, off, 32));
    if (lane == 0) sred[wave] = mx;
    __syncthreads();
    mx = sred[0];
#pragma unroll
    for (int w = 1; w < WAVES; ++w) mx = fmaxf(mx, sred[w]);

    float sum = 0.f;
#pragma unroll
    for (int u = 0; u < 8; ++u) { vals[u] = __expf(vals[u] - mx); sum += vals[u]; }
#pragma unroll
    for (int off = 16; off > 0; off >>= 1) sum += __shfl_xor(sum, off, 32);
    __syncthreads();
    if (lane == 0) sred[wave] = sum;
    __syncthreads();
    sum = 0.f;
#pragma unroll
    for (int w = 0; w < WAVES; ++w) sum += sred[w];

    const float inv = 1.0f / sum;
    float* arow = att + rowBase;
#pragma unroll
    for (int u = 0; u < 8; ++u) arow[tid + u * 256] = vals[u] * inv;
}

// ---------------------------------------------------------------------------
extern "C" void kernel_launch(void* const* d_in, const int* in_sizes, int n_in,
                              void* d_out, int out_size, void* d_ws, size_t ws_size,
                              hipStream_t stream)
{
    (void)in_sizes; (void)n_in; (void)out_size; (void)ws_size;
    const int B = 8, N = 2048, Fin = 256, Fout = 256;

    const float* h   = (const float*)d_in[0];   // (B, N, Fin)
    const int*   adj = (const int*)  d_in[1];   // (B, N, N)
    const float* W   = (const float*)d_in[2];   // (Fin, Fout)
    const float* a   = (const float*)d_in[3];   // (2*Fout, 1)

    float* out_h   = (float*)d_out;                       // elu(h') : B*N*Fout
    float* out_att = out_h + (long)B * N * Fout;          // attention: B*N*N

    float* Wh = (float*)d_ws;                             // B*N*Fout
    float* e1 = Wh + (long)B * N * Fout;                  // B*N
    float* e2 = e1 + (long)B * N;                         // B*N

    // 1) Wh = h @ W  (flatten batch: M = B*N)
    gat_gemm_wmma<<<dim3((B * N) / (16 * WAVES), 1), 256, 0, stream>>>(
        h, W, Wh, B * N, Fin, Fin, Fout, Fout, 0, 0, 0, 0);

    // 2) e1, e2
    gat_edge<<<dim3((B * N) / WAVES), 256, 0, stream>>>(Wh, a, e1, e2, Fout);

    // 3) attention rows -> d_out (second output)
    gat_softmax<<<dim3(N, B), 256, 0, stream>>>(adj, e1, e2, out_att, N);

    // 4) h' = elu(attention @ Wh) -> d_out (first output)
    gat_gemm_wmma<<<dim3(N / (16 * WAVES), B), 256, 0, stream>>>(
        out_att, Wh, out_h, N, N, N, Fout, Fout,
        (long)N * N, (long)N * Fout, (long)N * Fout, 1);
}